// GetWallDistPredictive_42820823941553
// MI455X (gfx1250) — compile-verified
//
#include <hip/hip_runtime.h>
#include <hip/hip_bf16.h>
#include <cstdint>

#define B_EV 32768
#define NP 253
#define N_ITER 50
#define TOL 1.0e-4f
#define INV2PI 0.15915494309189535f
#define TWOPI 6.283185307179586f

// Workspace layout (float offsets). Needs ~32 MB of ws.
#define WS_STEP 0
#define WS_SIN 1
#define WS_COS 2
#define WS_FLAG 3                 // int
#define WS_TE 16                  // float[B_EV]
#define WS_TP4 32784              // float4[NP]: {r^2+|p|^2, -2r*px, -2r*py, theta_pmt} (16B aligned)
#define WS_PXY (WS_TP4 + 4*NP)    // float2[NP]: {px, py}
#define WS_THETA 65536            // float[B_EV*NP]

__device__ __forceinline__ float hsin(float x) { return __builtin_amdgcn_sinf(x * INV2PI); }
__device__ __forceinline__ float hcos(float x) { return __builtin_amdgcn_cosf(x * INV2PI); }

// ---------- init kernels ----------
__global__ void k_init_scalars(float* ws) {
    ws[WS_STEP] = 0.1f;
    ws[WS_SIN]  = hsin(0.1f);
    ws[WS_COS]  = hcos(0.1f);
    ((int*)ws)[WS_FLAG] = 0;
}

__global__ void k_init_te(const float* __restrict__ X, float* __restrict__ ws) {
    int i = blockIdx.x * blockDim.x + threadIdx.x;
    if (i < B_EV) ws[WS_TE + i] = atan2f(X[2*i + 1], X[2*i]);
}

__global__ void k_init_pmt(const float* __restrict__ P, const float* __restrict__ R,
                           float* __restrict__ ws) {
    int p = threadIdx.x;
    if (p < NP) {
        float r  = *R;
        float px = P[2*p], py = P[2*p + 1];
        float4 v;
        v.x = fmaf(r, r, fmaf(px, px, py * py));
        v.y = -2.0f * r * px;
        v.z = -2.0f * r * py;
        v.w = atan2f(py, px);
        ((float4*)(ws + WS_TP4))[p] = v;
        ((float2*)(ws + WS_PXY))[p] = make_float2(px, py);
    }
}

__global__ void k_init_theta(float* __restrict__ ws) {
    int p = threadIdx.x;
    int b = blockIdx.y;
    if (p >= NP) return;
    float te = ws[WS_TE + b];                               // uniform -> s_load
    float tp = ((const float4*)(ws + WS_TP4))[p].w;
    float d  = tp - te;
    float delta = fmaf(-TWOPI, rintf(d * INV2PI), d);       // wrap to (-pi, pi]
    ws[WS_THETA + (size_t)b * NP + p] = fmaf(0.5f, delta, te);
}

// ---------- one coordinate-descent iteration (global step) ----------
__global__ void __launch_bounds__(256)
k_iter(const float* __restrict__ X, const float* __restrict__ R, float* __restrict__ ws) {
    const int tid = threadIdx.x;
    const int b   = blockIdx.y;
    __shared__ float4 spmt[NP];
    __shared__ int blkmv;

    float step = ws[WS_STEP];
    if (step < TOL) return;          // uniform: "active" is false, iteration is a no-op
    if (tid == 0) blkmv = 0;

    // Async-copy the PMT constant table into LDS (CDNA5 ASYNCcnt path).
    if (tid < NP) {
        uint32_t lds = (uint32_t)(uintptr_t)&spmt[tid];
        const float4* g = ((const float4*)(ws + WS_TP4)) + tid;
        asm volatile("global_load_async_to_lds_b128 %0, %1, off"
                     :: "v"(lds), "v"(g) : "memory");
    }
    asm volatile("s_wait_asynccnt 0x0" ::: "memory");
    __syncthreads();

    if (tid < NP) {
        float sin_s = ws[WS_SIN];
        float cos_s = ws[WS_COS];
        float r = *R;
        float2 e = ((const float2*)X)[b];                   // uniform -> s_load
        float ce0  = fmaf(r, r, fmaf(e.x, e.x, e.y * e.y));
        float nce1 = -2.0f * r * e.x;
        float nce2 = -2.0f * r * e.y;

        float4 pc = spmt[tid];
        size_t idx = (size_t)b * NP + tid;
        float th = ws[WS_THETA + idx];

        // single hardware sincos; neighbors via angle addition with uniform step
        float c = hcos(th), s = hsin(th);
        float cp = fmaf(c, cos_s, -(s * sin_s));
        float sp = fmaf(s, cos_s,  (c * sin_s));
        float cm = fmaf(c, cos_s,  (s * sin_s));
        float sm = fmaf(s, cos_s, -(c * sin_s));

        auto L = [&](float cc, float ss) {
            float de2 = fmaxf(fmaf(ss, nce2, fmaf(cc, nce1, ce0)), 0.0f);
            float dp2 = fmaxf(fmaf(ss, pc.z, fmaf(cc, pc.y, pc.x)), 0.0f);
            return __builtin_amdgcn_sqrtf(de2) + __builtin_amdgcn_sqrtf(dp2);
        };
        float l0 = L(c, s), lp = L(cp, sp), lm = L(cm, sm);
        float mv = (lp < l0) ? step : ((lm < l0) ? -step : 0.0f);
        if (mv != 0.0f) {
            ws[WS_THETA + idx] = th + mv;
            blkmv = 1;                                      // benign race
        }
    }
    __syncthreads();
    if (tid == 0 && blkmv) {
        int* flagp = ((int*)ws) + WS_FLAG;
        if (__hip_atomic_load(flagp, __ATOMIC_RELAXED, __HIP_MEMORY_SCOPE_AGENT) == 0)
            __hip_atomic_fetch_or(flagp, 1, __ATOMIC_RELAXED, __HIP_MEMORY_SCOPE_AGENT);
    }
}

// step/flag update: global "any moved" semantics of the reference
__global__ void k_step(float* ws) {
    float step = ws[WS_STEP];
    int* flagp = ((int*)ws) + WS_FLAG;
    if (step >= TOL && *flagp == 0) step *= 0.5f;
    *flagp = 0;
    ws[WS_STEP] = step;
    ws[WS_SIN]  = hsin(step);
    ws[WS_COS]  = hcos(step);
}

// ---------- finalize: distances (subtraction form for accuracy) ----------
__global__ void __launch_bounds__(256)
k_final(const float* __restrict__ X, const float* __restrict__ R,
        const float* __restrict__ ws, float* __restrict__ out) {
    const int tid = threadIdx.x;
    const int b   = blockIdx.y;
    __shared__ float4 spmt[NP];
    __shared__ float2 sxy[NP + 1];

    if (tid < NP) {
        uint32_t lds0 = (uint32_t)(uintptr_t)&spmt[tid];
        const float4* g0 = ((const float4*)(ws + WS_TP4)) + tid;
        asm volatile("global_load_async_to_lds_b128 %0, %1, off"
                     :: "v"(lds0), "v"(g0) : "memory");
        uint32_t lds1 = (uint32_t)(uintptr_t)&sxy[tid];
        const float2* g1 = ((const float2*)(ws + WS_PXY)) + tid;
        asm volatile("global_load_async_to_lds_b64 %0, %1, off"
                     :: "v"(lds1), "v"(g1) : "memory");
    }
    asm volatile("s_wait_asynccnt 0x0" ::: "memory");
    __syncthreads();
    if (tid >= NP) return;

    float r  = *R;
    float2 e = ((const float2*)X)[b];
    size_t idx = (size_t)b * NP + tid;
    float th = ws[WS_THETA + idx];
    bool center = fmaf(e.x, e.x, e.y * e.y) < 1.0e-16f;     // sqrt(|e|) < 1e-8
    if (center) th = spmt[tid].w;                            // theta_pmt

    float wx = r * cosf(th);
    float wy = r * sinf(th);
    float2 p = sxy[tid];
    float dex = wx - e.x, dey = wy - e.y;
    float dpx = wx - p.x, dpy = wy - p.y;
    float de = sqrtf(fmaf(dex, dex, dey * dey));
    float dp = sqrtf(fmaf(dpx, dpx, dpy * dpy));
    ((float2*)out)[idx] = make_float2(de, dp);
}

extern "C" void kernel_launch(void* const* d_in, const int* in_sizes, int n_in,
                              void* d_out, int out_size, void* d_ws, size_t ws_size,
                              hipStream_t stream) {
    const float* X = (const float*)d_in[0];   // (B, 2)
    const float* P = (const float*)d_in[1];   // (253, 2)
    const float* R = (const float*)d_in[2];   // scalar tpc_r
    float* ws  = (float*)d_ws;
    float* out = (float*)d_out;

    dim3 gB(1, B_EV, 1);
    k_init_scalars<<<1, 1, 0, stream>>>(ws);
    k_init_te<<<B_EV / 256, 256, 0, stream>>>(X, ws);
    k_init_pmt<<<1, 256, 0, stream>>>(P, R, ws);
    k_init_theta<<<gB, 256, 0, stream>>>(ws);
    for (int it = 0; it < N_ITER; ++it) {
        k_iter<<<gB, 256, 0, stream>>>(X, R, ws);
        k_step<<<1, 1, 0, stream>>>(ws);
    }
    k_final<<<gB, 256, 0, stream>>>(X, R, ws, out);
}